// GNNLayer_62156766708258
// MI455X (gfx1250) — compile-verified
//
#include <hip/hip_runtime.h>
#include <cstdint>

#define HEADS 4
#define CH 24
#define CIN 96
#define COUT 96
#define NEG_SLOPE 0.2f
#define LN_EPS 1e-5f

// K-pair-interleaved W layout in LDS: Wp[k2][col][k&1], row stride padded to 224
// dwords so lane-halves (k2 vs k2+1) hit disjoint bank halves (224 % 64 == 32).
#define WP_STRIDE 224
#define LDS_DWORDS ((CIN / 2) * WP_STRIDE)   // 48 * 224 = 10752 dwords = 42 KB

typedef __attribute__((ext_vector_type(2))) float v2f;
typedef __attribute__((ext_vector_type(8))) float v8f;

__device__ __forceinline__ unsigned fkey(float f) {
  unsigned b = __float_as_uint(f);
  return (b & 0x80000000u) ? ~b : (b | 0x80000000u);   // order-preserving key, all keys > 0
}
__device__ __forceinline__ float funkey(unsigned u) {
  unsigned b = (u & 0x80000000u) ? (u ^ 0x80000000u) : ~u;
  return __uint_as_float(b);
}
__device__ __forceinline__ float leaky(float e) { return e > 0.f ? e : NEG_SLOPE * e; }

// ---------------- init: zero accumulator, softmax max-keys, softmax sums ----------------
__global__ void k_init(float* __restrict__ out, unsigned* __restrict__ mkey,
                       float* __restrict__ ssum, int n) {
  int i = blockIdx.x * blockDim.x + threadIdx.x;
  if (i < n * COUT) out[i] = 0.f;
  if (i < n * HEADS) { mkey[i] = 0u; ssum[i] = 0.f; }
}

// ---------------- GEMM: h = x @ W via v_wmma_f32_16x16x4_f32, W packed in LDS ----------------
__global__ __launch_bounds__(256) void k_gemm(const float* __restrict__ x,
                                              const float* __restrict__ W,
                                              float* __restrict__ h,
                                              int n, int n_strips) {
  __shared__ float Wl[LDS_DWORDS];                 // 42 KB, K-pair interleaved
  for (int i = threadIdx.x; i < CIN * COUT; i += 256) {
    int k = i / COUT, col = i % COUT;
    Wl[(k >> 1) * WP_STRIDE + col * 2 + (k & 1)] = W[i];
  }
  __syncthreads();

  int wave = threadIdx.x >> 5;
  int lane = threadIdx.x & 31;
  int strip = blockIdx.x * 8 + wave;
  if (strip >= n_strips) return;                   // wave-uniform exit (EXEC all-1 for WMMA)

  int row_base = strip * 16;
  int lhalf = lane >> 4;                           // 0: K=0/1,  1: K=2/3
  int lmod  = lane & 15;
  int row   = row_base + lmod;
  if (row >= n) row = n - 1;                       // clamp (dup rows, results discarded)

  v8f acc[6] = {};                                 // 16 x 96 strip = 6 tiles of 16x16
  const float* xrow = x + (size_t)row * CIN;
  for (int k = 0; k < CIN; k += 4) {
    int k0 = k + (lhalf << 1);
    v2f a = *(const v2f*)(xrow + k0);              // global_load_b64, 8B aligned
    const float* wp = Wl + (k0 >> 1) * WP_STRIDE + lmod * 2;
#pragma unroll
    for (int t = 0; t < 6; ++t) {
      v2f b = *(const v2f*)(wp + t * 32);          // single ds_load_b64, bank-conflict-free
      acc[t] = __builtin_amdgcn_wmma_f32_16x16x4_f32(
          /*neg_a=*/false, a, /*neg_b=*/false, b,
          /*c_mod=*/(short)0, acc[t], /*reuse_a=*/false, /*reuse_b=*/false);
    }
  }
#pragma unroll
  for (int t = 0; t < 6; ++t) {
    int col = t * 16 + lmod;
#pragma unroll
    for (int v = 0; v < 8; ++v) {
      int r = row_base + v + 8 * lhalf;
      if (r < n) h[(size_t)r * COUT + col] = acc[t][v];
    }
  }
}

// ---------------- per-node attention coefficients a_src, a_dst ----------------
__global__ void k_att(const float* __restrict__ h, const float* __restrict__ att_src,
                      const float* __restrict__ att_dst,
                      float* __restrict__ asrc, float* __restrict__ adst, int n) {
  int i = blockIdx.x * blockDim.x + threadIdx.x;
  if (i >= n * HEADS) return;
  int head = i & (HEADS - 1);
  int node = i >> 2;
  const float* hp = h + (size_t)node * COUT + head * CH;
  const float* as = att_src + head * CH;
  const float* ad = att_dst + head * CH;
  float s0 = 0.f, s1 = 0.f;
#pragma unroll
  for (int c = 0; c < CH; ++c) { float v = hp[c]; s0 += v * as[c]; s1 += v * ad[c]; }
  asrc[i] = s0;
  adst[i] = s1;
}

// ---------------- segment max via ordered-uint atomicMax ----------------
__global__ void k_edge_max(const int* __restrict__ ei, int E, int n,
                           const float* __restrict__ asrc, const float* __restrict__ adst,
                           unsigned* __restrict__ mkey) {
  int i = blockIdx.x * blockDim.x + threadIdx.x;
  int EN = E + n;                                  // real edges + self loops
  if (i >= EN) return;
  int s = (i < E) ? ei[i] : (i - E);
  int d = (i < E) ? ei[E + i] : (i - E);
#pragma unroll
  for (int hd = 0; hd < HEADS; ++hd) {
    float e = leaky(asrc[s * HEADS + hd] + adst[d * HEADS + hd]);
    atomicMax(&mkey[d * HEADS + hd], fkey(e));
  }
}

// ---------------- segment sum of exp(e - max) ----------------
__global__ void k_edge_sum(const int* __restrict__ ei, int E, int n,
                           const float* __restrict__ asrc, const float* __restrict__ adst,
                           const unsigned* __restrict__ mkey, float* __restrict__ ssum) {
  int i = blockIdx.x * blockDim.x + threadIdx.x;
  int EN = E + n;
  if (i >= EN) return;
  int s = (i < E) ? ei[i] : (i - E);
  int d = (i < E) ? ei[E + i] : (i - E);
#pragma unroll
  for (int hd = 0; hd < HEADS; ++hd) {
    float e = leaky(asrc[s * HEADS + hd] + adst[d * HEADS + hd]);
    float ex = __expf(e - funkey(mkey[d * HEADS + hd]));
    atomicAdd(&ssum[d * HEADS + hd], ex);
  }
}

// ---------------- wave-per-edge message scatter: out[dst] += alpha * h[src] ----------------
__global__ __launch_bounds__(256) void k_msg(const int* __restrict__ ei, int E, int n,
                                             const float* __restrict__ h,
                                             const float* __restrict__ asrc,
                                             const float* __restrict__ adst,
                                             const unsigned* __restrict__ mkey,
                                             const float* __restrict__ ssum,
                                             float* __restrict__ out) {
  int wave = (blockIdx.x * blockDim.x + threadIdx.x) >> 5;
  int lane = threadIdx.x & 31;
  int EN = E + n;
  if (wave >= EN) return;
  int s = (wave < E) ? ei[wave] : (wave - E);
  int d = (wave < E) ? ei[E + wave] : (wave - E);
  float alpha[HEADS];
#pragma unroll
  for (int hd = 0; hd < HEADS; ++hd) {
    float e = leaky(asrc[s * HEADS + hd] + adst[d * HEADS + hd]);
    float ex = __expf(e - funkey(mkey[d * HEADS + hd]));
    alpha[hd] = ex / (ssum[d * HEADS + hd] + 1e-16f);
  }
  const float* hs = h + (size_t)s * COUT;
  float* od = out + (size_t)d * COUT;
#pragma unroll
  for (int j = 0; j < 3; ++j) {                    // 3 x 32 lanes = 96 channels, coalesced
    int c = lane + 32 * j;
    atomicAdd(&od[c], hs[c] * alpha[c / CH]);
  }
}

// ---------------- epilogue: +bias, +residual, LayerNorm (wave32 shuffle), ReLU ----------------
__global__ __launch_bounds__(256) void k_ln(const float* __restrict__ x,
                                            const float* __restrict__ bias,
                                            const float* __restrict__ gamma,
                                            const float* __restrict__ beta,
                                            float* __restrict__ out, int n) {
  int wave = (blockIdx.x * blockDim.x + threadIdx.x) >> 5;
  int lane = threadIdx.x & 31;
  if (wave >= n) return;
  float v[3];
  float lsum = 0.f;
#pragma unroll
  for (int j = 0; j < 3; ++j) {
    int c = lane + 32 * j;
    v[j] = out[(size_t)wave * COUT + c] + bias[c] + x[(size_t)wave * COUT + c];
    lsum += v[j];
  }
#pragma unroll
  for (int off = 16; off >= 1; off >>= 1) lsum += __shfl_xor(lsum, off, 32);
  float mu = lsum * (1.f / 96.f);
  float vs = 0.f;
#pragma unroll
  for (int j = 0; j < 3; ++j) { float dd = v[j] - mu; vs += dd * dd; }
#pragma unroll
  for (int off = 16; off >= 1; off >>= 1) vs += __shfl_xor(vs, off, 32);
  float rs = rsqrtf(vs * (1.f / 96.f) + LN_EPS);
#pragma unroll
  for (int j = 0; j < 3; ++j) {
    int c = lane + 32 * j;
    float o = (v[j] - mu) * rs * gamma[c] + beta[c];
    out[(size_t)wave * COUT + c] = fmaxf(o, 0.f);
  }
}

extern "C" void kernel_launch(void* const* d_in, const int* in_sizes, int n_in,
                              void* d_out, int out_size, void* d_ws, size_t ws_size,
                              hipStream_t stream) {
  const float* x       = (const float*)d_in[0];
  const int*   ei      = (const int*)d_in[1];
  const float* W       = (const float*)d_in[2];
  const float* att_src = (const float*)d_in[3];
  const float* att_dst = (const float*)d_in[4];
  const float* bias    = (const float*)d_in[5];
  const float* gamma   = (const float*)d_in[6];
  const float* beta    = (const float*)d_in[7];
  int n = in_sizes[0] / CIN;
  int E = in_sizes[1] / 2;
  float* out = (float*)d_out;

  // workspace layout
  char* ws = (char*)d_ws;
  float*    h    = (float*)ws;    ws += (size_t)n * COUT  * sizeof(float);
  float*    asrc = (float*)ws;    ws += (size_t)n * HEADS * sizeof(float);
  float*    adst = (float*)ws;    ws += (size_t)n * HEADS * sizeof(float);
  unsigned* mkey = (unsigned*)ws; ws += (size_t)n * HEADS * sizeof(unsigned);
  float*    ssum = (float*)ws;

  int EN = E + n;
  int strips = (n + 15) / 16;

  k_init<<<(n * COUT + 255) / 256, 256, 0, stream>>>(out, mkey, ssum, n);
  k_gemm<<<(strips + 7) / 8, 256, 0, stream>>>(x, W, h, n, strips);
  k_att<<<(n * HEADS + 255) / 256, 256, 0, stream>>>(h, att_src, att_dst, asrc, adst, n);
  k_edge_max<<<(EN + 255) / 256, 256, 0, stream>>>(ei, E, n, asrc, adst, mkey);
  k_edge_sum<<<(EN + 255) / 256, 256, 0, stream>>>(ei, E, n, asrc, adst, mkey, ssum);
  k_msg<<<(EN + 7) / 8, 256, 0, stream>>>(ei, E, n, h, asrc, adst, mkey, ssum, out);
  k_ln<<<(n + 7) / 8, 256, 0, stream>>>(x, bias, gamma, beta, out, n);
}